// GAT_62130996904395
// MI455X (gfx1250) — compile-verified
//
#include <hip/hip_runtime.h>
#include <hip/hip_bf16.h>

typedef float v2f __attribute__((ext_vector_type(2)));
typedef float v8f __attribute__((ext_vector_type(8)));

#define NEG_SLOPE 0.2f

// float atomic max via signed/unsigned integer trick
__device__ __forceinline__ void atomic_max_f32(float* addr, float val) {
  if (val >= 0.0f) {
    atomicMax((int*)addr, __float_as_int(val));
  } else {
    atomicMin((unsigned int*)addr, __float_as_uint(val));
  }
}

// ---------------------------------------------------------------------------
// GEMM D[M,Ncols] = A[M,K] * B[K,Ncols] using V_WMMA_F32_16X16X4_F32.
// One wave per 16x16 output tile; blockDim/32 waves cover Ncols/16 tiles.
// A fragment (16x4):  lanes 0-15 -> M=lane, K = k0+0 / k0+1
//                     lanes 16-31 -> M=lane-16, K = k0+2 / k0+3
// B fragment (4x16):  lanes 0-15 -> N=lane, K = k0+0 / k0+1 (VGPR0/1)
//                     lanes 16-31 -> N=lane-16, K = k0+2 / k0+3
// C/D (16x16): VGPR r: lanes 0-15 -> M=r, lanes 16-31 -> M=r+8, N=lane&15
// ---------------------------------------------------------------------------
__global__ void gemm_wmma_f32(const float* __restrict__ A,
                              const float* __restrict__ B,
                              float* __restrict__ D,
                              int M, int K, int Ncols) {
  const int lane = threadIdx.x & 31;
  const int wave = threadIdx.x >> 5;
  const int m0 = blockIdx.x * 16;
  const int n0 = wave * 16;
  if (m0 >= M || n0 >= Ncols) return;  // uniform per wave

  const int row = m0 + (lane & 15);
  const int col = n0 + (lane & 15);
  const int klo = (lane >> 4) << 1;  // 0 for lanes 0-15, 2 for lanes 16-31

  v8f acc = {};
  for (int k0 = 0; k0 < K; k0 += 4) {
    v2f a = *(const v2f*)(A + (size_t)row * K + k0 + klo);
    v2f b;
    b.x = B[(size_t)(k0 + klo) * Ncols + col];
    b.y = B[(size_t)(k0 + klo + 1) * Ncols + col];
    acc = __builtin_amdgcn_wmma_f32_16x16x4_f32(
        /*neg_a=*/false, a, /*neg_b=*/false, b,
        /*c_mod=*/(short)0, acc, /*reuse_a=*/false, /*reuse_b=*/false);
  }

  const int rbase = m0 + ((lane >> 4) << 3);  // +8 for upper half-wave
#pragma unroll
  for (int r = 0; r < 8; ++r) {
    D[(size_t)(rbase + r) * Ncols + (n0 + (lane & 15))] = acc[r];
  }
}

// ---------------------------------------------------------------------------
// Per-node attention scores: a_src[n,h] = sum_c h[n,h*64+c]*att_src[h*64+c]
// One wave per node. C is fixed to 64 (head = channel>>6).
// ---------------------------------------------------------------------------
__global__ void attn_kernel(const float* __restrict__ h,
                            const float* __restrict__ as,
                            const float* __restrict__ ad,
                            float* __restrict__ osrc, float* __restrict__ odst,
                            int N, int HC, int H) {
  const int node = blockIdx.x * (blockDim.x >> 5) + (threadIdx.x >> 5);
  if (node >= N) return;
  const int lane = threadIdx.x & 31;
  float s0 = 0.f, s1 = 0.f, d0 = 0.f, d1 = 0.f;
  for (int c = lane; c < HC; c += 32) {
    const float v = h[(size_t)node * HC + c];
    const float vs = v * as[c];
    const float vd = v * ad[c];
    if (c < 64) { s0 += vs; d0 += vd; } else { s1 += vs; d1 += vd; }
  }
#pragma unroll
  for (int off = 16; off; off >>= 1) {
    s0 += __shfl_xor(s0, off, 32);
    s1 += __shfl_xor(s1, off, 32);
    d0 += __shfl_xor(d0, off, 32);
    d1 += __shfl_xor(d1, off, 32);
  }
  if (lane == 0) {
    osrc[(size_t)node * H] = s0;
    odst[(size_t)node * H] = d0;
    if (H > 1) {
      osrc[(size_t)node * H + 1] = s1;
      odst[(size_t)node * H + 1] = d1;
    }
  }
}

// Initialize accumulators: out1 = b1 (broadcast), d_out = b2, maxima = -1e30,
// denominators = 0.
__global__ void init_kernel(float* __restrict__ out1, const float* __restrict__ b1,
                            float* __restrict__ dout, const float* __restrict__ b2,
                            float* __restrict__ m1, float* __restrict__ den1,
                            float* __restrict__ m2, float* __restrict__ den2,
                            int N) {
  const int i = blockIdx.x * blockDim.x + threadIdx.x;
  if (i < N * 128) out1[i] = b1[i & 127];
  if (i < N * 64) dout[i] = b2[i & 63];
  if (i < N * 2) { m1[i] = -1e30f; den1[i] = 0.f; }
  if (i < N)     { m2[i] = -1e30f; den2[i] = 0.f; }
}

// Pass 1: e = leaky_relu(a_src[src] + a_dst[dst]); atomic max per dst segment.
__global__ void edge_logits(const int* __restrict__ esrc, const int* __restrict__ edst,
                            int Ecnt, int N,
                            const float* __restrict__ a_src, const float* __restrict__ a_dst,
                            int H, float* __restrict__ ebuf, float* __restrict__ m) {
  const int e = blockIdx.x * blockDim.x + threadIdx.x;
  const int total = Ecnt + N;
  if (e >= total) return;
  const int s = (e < Ecnt) ? esrc[e] : (e - Ecnt);  // self-loops appended
  const int d = (e < Ecnt) ? edst[e] : (e - Ecnt);
  for (int hh = 0; hh < H; ++hh) {
    float v = a_src[(size_t)s * H + hh] + a_dst[(size_t)d * H + hh];
    v = (v > 0.f) ? v : NEG_SLOPE * v;
    ebuf[(size_t)e * H + hh] = v;
    atomic_max_f32(&m[(size_t)d * H + hh], v);
  }
}

// Pass 2: ex = exp(e - m[dst]); denom[dst] += ex; store ex in place.
__global__ void edge_exp_sum(const int* __restrict__ edst, int Ecnt, int N, int H,
                             float* __restrict__ ebuf, const float* __restrict__ m,
                             float* __restrict__ denom) {
  const int e = blockIdx.x * blockDim.x + threadIdx.x;
  const int total = Ecnt + N;
  if (e >= total) return;
  const int d = (e < Ecnt) ? edst[e] : (e - Ecnt);
  for (int hh = 0; hh < H; ++hh) {
    const float ex = __expf(ebuf[(size_t)e * H + hh] - m[(size_t)d * H + hh]);
    ebuf[(size_t)e * H + hh] = ex;
    unsafeAtomicAdd(&denom[(size_t)d * H + hh], ex);
  }
}

// Pass 3: out[dst] += h[src] * alpha. One wave per edge; vectorized gather,
// hardware f32 atomics for the scatter.
template <int HC, int H>
__global__ void edge_aggregate(const int* __restrict__ esrc, const int* __restrict__ edst,
                               int Ecnt, int N,
                               const float* __restrict__ ebuf, const float* __restrict__ denom,
                               const float* __restrict__ hfeat, float* __restrict__ out) {
  const int e = blockIdx.x * (blockDim.x >> 5) + (threadIdx.x >> 5);
  const int total = Ecnt + N;
  if (e >= total) return;
  const int lane = threadIdx.x & 31;
  const int s = (e < Ecnt) ? esrc[e] : (e - Ecnt);
  const int d = (e < Ecnt) ? edst[e] : (e - Ecnt);
  constexpr int PER = HC / 32;  // 4 (layer1) or 2 (layer2) channels per lane
  const int c0 = lane * PER;
  const int head = (H == 1) ? 0 : (c0 >> 6);
  const float alpha =
      ebuf[(size_t)e * H + head] / (denom[(size_t)d * H + head] + 1e-16f);
  const float* hp = hfeat + (size_t)s * HC + c0;
  float* op = out + (size_t)d * HC + c0;
  if (PER == 4) {
    const float4 hv = *(const float4*)hp;
    unsafeAtomicAdd(&op[0], hv.x * alpha);
    unsafeAtomicAdd(&op[1], hv.y * alpha);
    unsafeAtomicAdd(&op[2], hv.z * alpha);
    unsafeAtomicAdd(&op[3], hv.w * alpha);
  } else {
    const float2 hv = *(const float2*)hp;
    unsafeAtomicAdd(&op[0], hv.x * alpha);
    unsafeAtomicAdd(&op[1], hv.y * alpha);
  }
}

__global__ void relu_kernel(float* __restrict__ x, int n) {
  const int i = blockIdx.x * blockDim.x + threadIdx.x;
  if (i < n) x[i] = fmaxf(x[i], 0.f);
}

extern "C" void kernel_launch(void* const* d_in, const int* in_sizes, int n_in,
                              void* d_out, int out_size, void* d_ws, size_t ws_size,
                              hipStream_t stream) {
  const float* X   = (const float*)d_in[0];
  const int*   E   = (const int*)d_in[1];   // int32 per harness convention
  const float* W1  = (const float*)d_in[2];
  const float* as1 = (const float*)d_in[3];
  const float* ad1 = (const float*)d_in[4];
  const float* b1  = (const float*)d_in[5];
  const float* W2  = (const float*)d_in[6];
  const float* as2 = (const float*)d_in[7];
  const float* ad2 = (const float*)d_in[8];
  const float* b2  = (const float*)d_in[9];
  float* out = (float*)d_out;

  const int N    = in_sizes[0] / 256;  // F_IN = 256
  const int Ecnt = in_sizes[1] / 2;
  const int Etot = Ecnt + N;           // with self-loops
  const int* esrc = E;
  const int* edst = E + Ecnt;

  // Workspace carve-out (floats). h2 aliases h1 (h1 dead after layer-1 agg).
  float* ws = (float*)d_ws;
  size_t off = 0;
  float* h1     = ws + off; off += (size_t)N * 128;
  float* out1   = ws + off; off += (size_t)N * 128;
  float* a_src1 = ws + off; off += (size_t)N * 2;
  float* a_dst1 = ws + off; off += (size_t)N * 2;
  float* m1     = ws + off; off += (size_t)N * 2;
  float* den1   = ws + off; off += (size_t)N * 2;
  float* a_src2 = ws + off; off += (size_t)N;
  float* a_dst2 = ws + off; off += (size_t)N;
  float* m2     = ws + off; off += (size_t)N;
  float* den2   = ws + off; off += (size_t)N;
  float* ebuf   = ws + off; off += (size_t)Etot * 2;
  float* h2 = h1;

  const int mtiles   = (N + 15) / 16;
  const int nodeBlk  = (N + 7) / 8;            // 8 waves/block, 1 node/wave
  const int elemBlk  = (N * 128 + 255) / 256;
  const int edgeBlk  = (Etot + 255) / 256;
  const int edgeWBlk = (Etot + 7) / 8;         // 8 waves/block, 1 edge/wave

  // ---- Layer 1 ----
  gemm_wmma_f32<<<mtiles, 256, 0, stream>>>(X, W1, h1, N, 256, 128);
  attn_kernel<<<nodeBlk, 256, 0, stream>>>(h1, as1, ad1, a_src1, a_dst1, N, 128, 2);
  init_kernel<<<elemBlk, 256, 0, stream>>>(out1, b1, out, b2, m1, den1, m2, den2, N);
  edge_logits<<<edgeBlk, 256, 0, stream>>>(esrc, edst, Ecnt, N, a_src1, a_dst1, 2, ebuf, m1);
  edge_exp_sum<<<edgeBlk, 256, 0, stream>>>(edst, Ecnt, N, 2, ebuf, m1, den1);
  edge_aggregate<128, 2><<<edgeWBlk, 256, 0, stream>>>(esrc, edst, Ecnt, N, ebuf, den1, h1, out1);
  relu_kernel<<<elemBlk, 256, 0, stream>>>(out1, N * 128);

  // ---- Layer 2 ----
  gemm_wmma_f32<<<mtiles, 128, 0, stream>>>(out1, W2, h2, N, 128, 64);
  attn_kernel<<<nodeBlk, 256, 0, stream>>>(h2, as2, ad2, a_src2, a_dst2, N, 64, 1);
  edge_logits<<<edgeBlk, 256, 0, stream>>>(esrc, edst, Ecnt, N, a_src2, a_dst2, 1, ebuf, m2);
  edge_exp_sum<<<edgeBlk, 256, 0, stream>>>(edst, Ecnt, N, 1, ebuf, m2, den2);
  edge_aggregate<64, 1><<<edgeWBlk, 256, 0, stream>>>(esrc, edst, Ecnt, N, ebuf, den2, h2, out);
}